// VolumeSDFRenderer_78374563217919
// MI455X (gfx1250) — compile-verified
//
#include <hip/hip_runtime.h>
#include <stdint.h>

// VolSDF volume render, MI455X (gfx1250, wave32).
// Memory-bound (~160 MB single-pass => ~7us floor @ 23.3 TB/s).
// - one wave32 per ray (N=128 samples, 4 per lane), EXEC always full
// - color stream (96 MB) staged via CDNA5 async global->LDS DMA (ASYNCcnt),
//   overlapped with density + wave-scan compute, consumed with ds_load_b128
// - all streaming global loads marked non-temporal (single-pass data,
//   don't rinse WGP$/L2)
// - exclusive cumsum: 4-wide serial local prefix + 5-step wave shuffle scan
// - RGB reduction: 5-step xor butterfly, lane 0 stores color + zero geometry

#define RENDER_ALPHA   10.0f
#define RENDER_INVBETA 20.0f      // 1/0.05
#define FAR_DELTA      1e10f

typedef float v4f __attribute__((ext_vector_type(4)));

__device__ __forceinline__ float sdf_density(float dist) {
    // s = -dist; density = alpha*0.5*exp(s/beta)        if s<=0
    //                     alpha*(1-0.5*exp(-s/beta))    otherwise
    // Both branches use exp(-|s|/beta); select the affine form.
    float s = -dist;
    float e = expf(-fabsf(s) * RENDER_INVBETA);
    return (s <= 0.0f) ? (RENDER_ALPHA * 0.5f * e)
                       : (RENDER_ALPHA * (1.0f - 0.5f * e));
}

__launch_bounds__(256, 4)
__global__ void volsdf_render_kernel(const float* __restrict__ distance,
                                     const float* __restrict__ color,
                                     const float* __restrict__ depth,
                                     float* __restrict__ out,
                                     int M) {
    constexpr int N = 128;           // samples per ray
    constexpr int WAVES = 8;         // rays per block
    __shared__ __align__(16) float scol[WAVES * N * 3];   // 12 KB color staging

    const int lane = threadIdx.x & 31;
    const int wave = threadIdx.x >> 5;
    const int ray  = blockIdx.x * WAVES + wave;           // wave-uniform
    if (ray >= M) return;

    // ------------------------------------------------------------------
    // 1) Kick off async DMA of this ray's color (128*3 floats = 1536 B)
    //    into this wave's private LDS region. 32 lanes x 16 B x 3 rounds.
    //    Non-temporal: the color stream is touched exactly once.
    // ------------------------------------------------------------------
    const uint32_t lds_wave =
        (uint32_t)(uintptr_t)scol + (uint32_t)(wave * N * 3 * sizeof(float));
    const float* cbase = color + (size_t)ray * (N * 3);
#pragma unroll
    for (int k = 0; k < 3; ++k) {
        uint32_t loff = lds_wave + (uint32_t)((lane + 32 * k) * 16);
        uint32_t goff = (uint32_t)((lane + 32 * k) * 16);
        asm volatile("global_load_async_to_lds_b128 %0, %1, %2 th:TH_LOAD_NT"
                     :: "v"(loff), "v"(goff), "s"(cbase)
                     : "memory");
    }

    // ------------------------------------------------------------------
    // 2) Meanwhile: load distance & depth (b128 NT, coalesced), build
    //    deltas, densities, and the exclusive prefix sum of sigma*delta.
    //    Lane l owns samples 4l..4l+3.
    // ------------------------------------------------------------------
    const int j0 = lane * 4;
    const v4f dist4 =
        __builtin_nontemporal_load((const v4f*)(distance + (size_t)ray * N + j0));
    const v4f dep4 =
        __builtin_nontemporal_load((const v4f*)(depth + (size_t)ray * N + j0));

    const float nextd  = __shfl_down(dep4.x, 1, 32);      // depth[4l+4]
    const float delta0 = dep4.y - dep4.x;
    const float delta1 = dep4.z - dep4.y;
    const float delta2 = dep4.w - dep4.z;
    const float delta3 = (lane == 31) ? FAR_DELTA : (nextd - dep4.w);

    const float sd0 = sdf_density(dist4.x) * delta0;
    const float sd1 = sdf_density(dist4.y) * delta1;
    const float sd2 = sdf_density(dist4.z) * delta2;
    const float sd3 = sdf_density(dist4.w) * delta3;

    // local inclusive prefix within the lane's 4 samples
    const float p1 = sd0 + sd1;
    const float p2 = p1 + sd2;
    const float p3 = p2 + sd3;

    // wave32 inclusive scan of per-lane sums -> exclusive lane offset
    float x = p3;
#pragma unroll
    for (int off = 1; off < 32; off <<= 1) {
        float y = __shfl_up(x, off, 32);
        if (lane >= off) x += y;
    }
    const float excl = x - p3;     // sum of sigma*delta for samples < 4l

    // transmittance T_i = exp(-exclusive_cumsum_i), weights
    const float T0 = expf(-excl);
    const float T1 = expf(-(excl + sd0));
    const float T2 = expf(-(excl + p1));
    const float T3 = expf(-(excl + p2));
    const float w0 = T0 * (1.0f - expf(-sd0));
    const float w1 = T1 * (1.0f - expf(-sd1));
    const float w2 = T2 * (1.0f - expf(-sd2));
    const float w3 = T3 * (1.0f - expf(-sd3));

    // ------------------------------------------------------------------
    // 3) Wait for the async color DMA, read 12 floats (3 x b128) from LDS.
    //    Lane l's 48-byte slice holds colors of samples 4l..4l+3.
    // ------------------------------------------------------------------
    asm volatile("s_wait_asynccnt 0x0" ::: "memory");

    const float4* cl = (const float4*)(scol + wave * N * 3 + lane * 12);
    const float4 c0 = cl[0];
    const float4 c1 = cl[1];
    const float4 c2 = cl[2];
    // sample k color: k=0 -> (c0.x,c0.y,c0.z), k=1 -> (c0.w,c1.x,c1.y),
    //                 k=2 -> (c1.z,c1.w,c2.x), k=3 -> (c2.y,c2.z,c2.w)
    float r = w0 * c0.x + w1 * c0.w + w2 * c1.z + w3 * c2.y;
    float g = w0 * c0.y + w1 * c1.x + w2 * c1.w + w3 * c2.z;
    float b = w0 * c0.z + w1 * c1.y + w2 * c2.x + w3 * c2.w;

    // ------------------------------------------------------------------
    // 4) Wave reduction over the 32 lanes (128 samples), lane 0 stores.
    // ------------------------------------------------------------------
#pragma unroll
    for (int off = 16; off > 0; off >>= 1) {
        r += __shfl_xor(r, off, 32);
        g += __shfl_xor(g, off, 32);
        b += __shfl_xor(b, off, 32);
    }

    if (lane == 0) {
        float* oc = out + (size_t)ray * 3;
        oc[0] = r; oc[1] = g; oc[2] = b;
        // geometry output stays zeros (light_dir=None path)
        float* og = out + (size_t)M * 3 + (size_t)ray * 3;
        og[0] = 0.0f; og[1] = 0.0f; og[2] = 0.0f;
    }
}

extern "C" void kernel_launch(void* const* d_in, const int* in_sizes, int n_in,
                              void* d_out, int out_size, void* d_ws, size_t ws_size,
                              hipStream_t stream) {
    const float* distance = (const float*)d_in[0];   // [M, N, 1] f32
    const float* color    = (const float*)d_in[1];   // [M, N, 3] f32
    const float* depth    = (const float*)d_in[2];   // [M, N]    f32
    float* out = (float*)d_out;                      // [M,3] color ++ [M,3] geometry

    const int N = 128;
    const int M = in_sizes[2] / N;                   // 65536
    const int blocks = (M + 7) / 8;                  // 8 rays (waves) per block

    volsdf_render_kernel<<<blocks, 256, 0, stream>>>(distance, color, depth, out, M);
}